// ChamferLoss_11776800325759
// MI455X (gfx1250) — compile-verified
//
#include <hip/hip_runtime.h>

// Chamfer distance on MI455X (gfx1250, wave32).
// Cross-term s.d computed with V_WMMA_F32_16X16X4_F32 (16x16 dot tile / instr).

typedef __attribute__((ext_vector_type(2))) float v2f;
typedef __attribute__((ext_vector_type(8))) float v8f;

#define CH_B 8
#define CH_N 4096
#define CH_M 4096
#define WAVES_PER_BLOCK 8   // 256 threads = 8 wave32

// One wave per 16-row tile of `src`; loops over all M `dst` points in
// 16-column steps, one WMMA per step. Writes sum of the 16 row-mins to
// partial[b*(N/16) + tile].
__global__ __launch_bounds__(256) void chamfer_rowmin_kernel(
    const float* __restrict__ src,   // [B, N, 3]
    const float* __restrict__ dst,   // [B, M, 3]
    float* __restrict__ partial,     // [B * N/16]
    int N, int M)
{
    const int lane = threadIdx.x & 31;
    const int wave = threadIdx.x >> 5;
    const int tilesPerBatch  = N >> 4;                       // N/16
    const int blocksPerBatch = tilesPerBatch / WAVES_PER_BLOCK;
    const int b           = blockIdx.x / blocksPerBatch;
    const int tileInBatch = (blockIdx.x % blocksPerBatch) * WAVES_PER_BLOCK + wave;
    const int rowBase     = tileInBatch << 4;

    const float* __restrict__ srcB = src + (size_t)b * N * 3;
    const float* __restrict__ dstB = dst + (size_t)b * M * 3;

    const int hi = lane >> 4;    // 0 => K=0..1 slice, 1 => K=2..3 slice
    const int mr = lane & 15;    // row (A) / col (B) index within tile

    // ---- A matrix (16x4 f32, K=3 padded with 0). ISA 32-bit A layout:
    // lanes 0-15: M=lane, VGPR0=K0, VGPR1=K1; lanes 16-31: M=lane-16, K2,K3.
    const float* sp = srcB + (size_t)(rowBase + mr) * 3;
    const float s0 = sp[0], s1 = sp[1], s2 = sp[2];
    v2f a;
    a.x = hi ? s2 : s0;
    a.y = hi ? 0.0f : s1;

    // ---- Row norms for the 8 rows this lane's C/D slice covers
    // (C/D layout: lanes 0-15 -> rows r, lanes 16-31 -> rows r+8).
    float snorm[8];
#pragma unroll
    for (int r = 0; r < 8; ++r) {
        const float* q = srcB + (size_t)(rowBase + r + 8 * hi) * 3;
        const float q0 = q[0], q1 = q[1], q2 = q[2];
        snorm[r] = q0 * q0 + q1 * q1 + q2 * q2;
    }

    float rmin[8];
#pragma unroll
    for (int r = 0; r < 8; ++r) rmin[r] = 3.4e38f;

    for (int cb = 0; cb < M; cb += 16) {
        const float* dp = dstB + (size_t)(cb + mr) * 3;
        // Prefetch next column tile (global_prefetch_b8).
        __builtin_prefetch((const void*)(dp + 48), 0, 1);
        const float d0 = dp[0], d1 = dp[1], d2 = dp[2];
        const float dnorm = d0 * d0 + d1 * d1 + d2 * d2;

        // B matrix (4x16 f32): mirrored striping of the A layout —
        // lanes 0-15 carry K0/K1 of column `lane`, lanes 16-31 carry K2/K3.
        v2f bm;
        bm.x = hi ? d2 : d0;
        bm.y = hi ? 0.0f : d1;

        v8f c = {};
        // D = A x B (+0): 16x16 tile of dot products in one instruction.
        c = __builtin_amdgcn_wmma_f32_16x16x4_f32(
            /*neg_a=*/false, a, /*neg_b=*/false, bm,
            /*c_mod=*/(short)0, c, /*reuse_a=*/false, /*reuse_b=*/false);

#pragma unroll
        for (int r = 0; r < 8; ++r) {
            const float dist = __builtin_fmaf(-2.0f, c[r], dnorm); // ||d||^2 - 2 s.d
            rmin[r] = fminf(rmin[r], dist);
        }
    }

    // Min over the 16 columns held by each 16-lane half (xor 1,2,4,8 stays
    // inside each half of the wave32), then add the row norm ||s||^2.
#pragma unroll
    for (int r = 0; r < 8; ++r) {
        float v = rmin[r];
        v = fminf(v, __shfl_xor(v, 1, 32));
        v = fminf(v, __shfl_xor(v, 2, 32));
        v = fminf(v, __shfl_xor(v, 4, 32));
        v = fminf(v, __shfl_xor(v, 8, 32));
        rmin[r] = v + snorm[r];
    }
    float s = 0.0f;
#pragma unroll
    for (int r = 0; r < 8; ++r) s += rmin[r];
    s += __shfl_xor(s, 16, 32);   // rows 0-7 half + rows 8-15 half

    if (lane == 0) partial[(size_t)b * tilesPerBatch + tileInBatch] = s;
}

// Deterministic fixed-order reduction of both passes' partials (no float
// atomics -> bit-identical across replays).
__global__ __launch_bounds__(256) void chamfer_final_kernel(
    const float* __restrict__ partial, float* __restrict__ out,
    int n0, int n1, float scale0, float scale1)
{
    __shared__ float l0[256];
    __shared__ float l1[256];
    const int t = threadIdx.x;
    float s0 = 0.0f, s1 = 0.0f;
    for (int i = t; i < n0; i += 256) s0 += partial[i];
    for (int i = t; i < n1; i += 256) s1 += partial[n0 + i];
    l0[t] = s0; l1[t] = s1;
    __syncthreads();
    for (int off = 128; off > 0; off >>= 1) {
        if (t < off) { l0[t] += l0[t + off]; l1[t] += l1[t + off]; }
        __syncthreads();
    }
    if (t == 0) out[0] = l0[0] * scale0 + l1[0] * scale1;
}

extern "C" void kernel_launch(void* const* d_in, const int* in_sizes, int n_in,
                              void* d_out, int out_size, void* d_ws, size_t ws_size,
                              hipStream_t stream) {
    (void)in_sizes; (void)n_in; (void)out_size; (void)ws_size;
    const float* f  = (const float*)d_in[0];   // [B,N,3]
    const float* f_ = (const float*)d_in[1];   // [B,M,3]
    float* out = (float*)d_out;
    float* wsf = (float*)d_ws;

    const int B = CH_B, N = CH_N, M = CH_M;
    const int tilesN = N >> 4;                    // 256 per batch
    const int tilesM = M >> 4;
    const int blocksP0 = B * tilesN / WAVES_PER_BLOCK;   // 256 blocks
    const int blocksP1 = B * tilesM / WAVES_PER_BLOCK;

    // Pass 0: for each f point, min distance to f_ (row mins).
    chamfer_rowmin_kernel<<<blocksP0, 256, 0, stream>>>(f, f_, wsf, N, M);
    // Pass 1: roles swapped.
    chamfer_rowmin_kernel<<<blocksP1, 256, 0, stream>>>(f_, f, wsf + (size_t)B * tilesN, M, N);
    // Final: mean over batches of (mean_n + mean_m).
    chamfer_final_kernel<<<1, 256, 0, stream>>>(
        wsf, out, B * tilesN, B * tilesM,
        1.0f / ((float)B * (float)N), 1.0f / ((float)B * (float)M));
}